// FasterRCNN_35794257445058
// MI455X (gfx1250) — compile-verified
//
#include <hip/hip_runtime.h>
#include <hip/hip_bf16.h>

#define NP 50000          // pred boxes per batch (divisible by 4)
#define NPV (NP / 4)      // 12500 float4 groups
#define NG 64
#define NB 8
#define POS_T 0.7f
#define NEG_T 0.3f

typedef float v4f __attribute__((ext_vector_type(4)));

#if __has_builtin(__builtin_amdgcn_global_load_async_to_lds_b32)
#define HAVE_ASYNC_LDS 1
#else
#define HAVE_ASYNC_LDS 0
#pragma message("gfx1250 async-to-LDS builtin NOT available; using plain LDS staging")
#endif

typedef __attribute__((address_space(1))) int* gptr_i32;
typedef __attribute__((address_space(3))) int* lptr_i32;

// ---------------------------------------------------------------------------
// Kernel 1: one thread per (batch, 4 consecutive pred boxes). gt boxes staged
// in LDS via gfx1250 async global->LDS. Per gt, each thread emits one b128
// iou store (RT: re-read by pass 2 from the 192MB L2) and one b128 pos-flag
// NT store; finally one b128 NT neg store. All stores are 16B/lane.
// ---------------------------------------------------------------------------
__global__ __launch_bounds__(256) void iou_main_kernel(
    const v4f* __restrict__ pred,      // (NB, NP, 4) cxcywh, one v4f per box
    const v4f* __restrict__ gt,        // (NB, NG, 4) cxcywh
    float* __restrict__ out_iou,       // (NB, NG, NP)
    float* __restrict__ out_pos,       // (NB, NG, NP)
    float* __restrict__ out_neg)       // (NB, NP)
{
    __shared__ v4f   rawgt[NG];
    __shared__ float gx0[NG], gy0[NG], gx1[NG], gy1[NG], gar[NG];

    const int t = threadIdx.x;
    const int b = blockIdx.y;
    const int v = blockIdx.x * 256 + t;      // float4-group index
    const bool act = v < NPV;
    const int p0 = v * 4;                    // first pred index of this thread

    // Issue the 4 pred-box loads early so they overlap LDS staging + barriers.
    v4f pb[4];
    if (act) {
        const v4f* pp = pred + (size_t)b * NP + p0;
        pb[0] = pp[0]; pb[1] = pp[1]; pb[2] = pp[2]; pb[3] = pp[3];
    } else {
        pb[0] = pb[1] = pb[2] = pb[3] = (v4f)0.f;
    }

#if HAVE_ASYNC_LDS
    {
        // 64 boxes * 16B = 1024B = 256 dwords: every thread asynchronously
        // copies one dword straight into LDS (no VGPR round trip).
        const int* gsrc = (const int*)(gt + (size_t)b * NG);
        __builtin_amdgcn_global_load_async_to_lds_b32(
            (gptr_i32)(gsrc + t),
            (lptr_i32)((int*)rawgt + t),
            /*offset=*/0, /*cpol=*/0);
#if __has_builtin(__builtin_amdgcn_s_wait_asynccnt)
        __builtin_amdgcn_s_wait_asynccnt(0);
#else
        asm volatile("s_wait_asynccnt 0" ::: "memory");
#endif
        __syncthreads();
    }
    if (t < NG) {
        v4f g = rawgt[t];
        float x0 = g.x - 0.5f * g.z, y0 = g.y - 0.5f * g.w;
        float x1 = g.x + 0.5f * g.z, y1 = g.y + 0.5f * g.w;
        gx0[t] = x0; gy0[t] = y0; gx1[t] = x1; gy1[t] = y1;
        gar[t] = (x1 - x0) * (y1 - y0);
    }
#else
    if (t < NG) {
        v4f g = gt[(size_t)b * NG + t];
        float x0 = g.x - 0.5f * g.z, y0 = g.y - 0.5f * g.w;
        float x1 = g.x + 0.5f * g.z, y1 = g.y + 0.5f * g.w;
        gx0[t] = x0; gy0[t] = y0; gx1[t] = x1; gy1[t] = y1;
        gar[t] = (x1 - x0) * (y1 - y0);
    }
#endif
    __syncthreads();

    if (!act) return;

    float px0[4], py0[4], px1[4], py1[4], parea[4];
#pragma unroll
    for (int k = 0; k < 4; ++k) {
        px0[k] = pb[k].x - 0.5f * pb[k].z;  py0[k] = pb[k].y - 0.5f * pb[k].w;
        px1[k] = pb[k].x + 0.5f * pb[k].z;  py1[k] = pb[k].y + 0.5f * pb[k].w;
        parea[k] = (px1[k] - px0[k]) * (py1[k] - py0[k]);
    }

    v4f m = (v4f)(-1.0f);
    const size_t base = (size_t)b * NG * NP + p0;

#pragma unroll 2
    for (int g = 0; g < NG; ++g) {
        // LDS reads: all 32 lanes hit the same address -> conflict-free broadcast.
        const float x0 = gx0[g], y0 = gy0[g], x1 = gx1[g], y1 = gy1[g], ga = gar[g];
        v4f iou4, pos4;
#pragma unroll
        for (int k = 0; k < 4; ++k) {
            float ltx = fmaxf(x0, px0[k]), lty = fmaxf(y0, py0[k]);
            float rbx = fminf(x1, px1[k]), rby = fminf(y1, py1[k]);
            float w = fmaxf(rbx - ltx, 0.f), h = fmaxf(rby - lty, 0.f);
            float inter = w * h;
            float uni = ga + parea[k] - inter;
            float iou = inter * __builtin_amdgcn_rcpf(uni);  // v_rcp_f32, ~1ulp
            iou4[k] = iou;
            pos4[k] = iou > POS_T ? 1.0f : 0.0f;
            m[k] = fmaxf(m[k], iou);
        }
        size_t o = base + (size_t)g * NP;                 // 16B-aligned, coalesced
        *(v4f*)(out_iou + o) = iou4;                      // RT b128: L2-resident for pass 2
        __builtin_nontemporal_store(pos4, (v4f*)(out_pos + o));  // NT b128: pure stream
    }
    v4f neg4;
#pragma unroll
    for (int k = 0; k < 4; ++k) neg4[k] = m[k] < NEG_T ? 1.0f : 0.0f;
    __builtin_nontemporal_store(neg4, (v4f*)(out_neg + (size_t)b * NP + p0));
}

// ---------------------------------------------------------------------------
// Kernel 2: per (batch, gt) argmax over 50000 ious (hot in L2, b128 reads),
// OR the onehot into pos_mask. wave32 shuffle reduction + LDS cross-wave
// combine. Ascending-order scalar checks keep jnp.argmax first-index ties.
// ---------------------------------------------------------------------------
__global__ __launch_bounds__(256) void argmax_onehot_kernel(
    const float* __restrict__ ious, float* __restrict__ out_pos)
{
    const int g = blockIdx.x;
    const int b = blockIdx.y;
    const int t = threadIdx.x;
    const v4f* rowv = (const v4f*)(ious + ((size_t)b * NG + g) * NP);

    float bv = -1.0f;
    int bi = 0;
    for (int i = t; i < NPV; i += 256) {     // ascending per thread
        v4f val = rowv[i];
        int p = i * 4;
        if (val.x > bv) { bv = val.x; bi = p; }
        if (val.y > bv) { bv = val.y; bi = p + 1; }
        if (val.z > bv) { bv = val.z; bi = p + 2; }
        if (val.w > bv) { bv = val.w; bi = p + 3; }
    }
    // wave32 butterfly reduction
    for (int off = 16; off > 0; off >>= 1) {
        float ov = __shfl_xor(bv, off, 32);
        int   oi = __shfl_xor(bi, off, 32);
        if (ov > bv || (ov == bv && oi < bi)) { bv = ov; bi = oi; }
    }
    __shared__ float sv[8];
    __shared__ int   si[8];
    const int wid = t >> 5, lane = t & 31;
    if (lane == 0) { sv[wid] = bv; si[wid] = bi; }
    __syncthreads();
    if (t == 0) {
        for (int w = 1; w < 8; ++w)
            if (sv[w] > bv || (sv[w] == bv && si[w] < bi)) { bv = sv[w]; bi = si[w]; }
        out_pos[((size_t)b * NG + g) * NP + bi] = 1.0f;
    }
}

extern "C" void kernel_launch(void* const* d_in, const int* in_sizes, int n_in,
                              void* d_out, int out_size, void* d_ws, size_t ws_size,
                              hipStream_t stream) {
    (void)in_sizes; (void)n_in; (void)out_size; (void)d_ws; (void)ws_size;

    const v4f* pred = (const v4f*)d_in[0];  // (8, 50000, 4) f32
    const v4f* gt   = (const v4f*)d_in[1];  // (8, 64, 4)    f32

    float* out     = (float*)d_out;
    float* out_iou = out;                               // 25,600,000 f32
    float* out_pos = out_iou + (size_t)NB * NG * NP;    // 25,600,000 f32
    float* out_neg = out_pos + (size_t)NB * NG * NP;    //    400,000 f32

    dim3 grid1((NPV + 255) / 256, NB);                  // 49 x 8 blocks
    iou_main_kernel<<<grid1, 256, 0, stream>>>(pred, gt, out_iou, out_pos, out_neg);

    dim3 grid2(NG, NB);                                 // 64 x 8 blocks
    argmax_onehot_kernel<<<grid2, 256, 0, stream>>>(out_iou, out_pos);
}